// EdgeClassifier_61108794688063
// MI455X (gfx1250) — compile-verified
//
#include <hip/hip_runtime.h>
#include <cstdint>

typedef __attribute__((ext_vector_type(2))) float v2f;
typedef __attribute__((ext_vector_type(8))) float v8f;
typedef __attribute__((ext_vector_type(4))) int   v4i;

typedef __attribute__((address_space(1))) v4i* gbl_v4i_ptr;
typedef __attribute__((address_space(3))) v4i* lds_v4i_ptr;

#define IN_CH 128
#define OUT_CH 64
#define TILE_M 16

// ---------------- degree / normalization ----------------
__global__ void k_init_deg(float* deg, int N) {
  int i = blockIdx.x * blockDim.x + threadIdx.x;
  if (i < N) deg[i] = 1.0f;                       // self-loop
}

__global__ void k_deg_edges(const int* __restrict__ ei, float* deg, int E) {
  int e = blockIdx.x * blockDim.x + threadIdx.x;
  if (e < E) atomicAdd(&deg[ei[E + e]], 1.0f);    // col = target
}

__global__ void k_dinv(float* deg, int N) {
  int i = blockIdx.x * blockDim.x + threadIdx.x;
  if (i < N) deg[i] = rsqrtf(deg[i]);             // deg >= 1 always
}

// ---------------- h = x @ W via V_WMMA_F32_16X16X4_F32 ----------------
// Block = 16-row M tile staged once into LDS (async copy), 4 waves each
// owning a 16-column N tile so the x-tile is read from global exactly once.
__global__ __launch_bounds__(128) void k_gemm_wmma(const float* __restrict__ x,
                                                   const float* __restrict__ W,
                                                   float* __restrict__ h) {
  __shared__ float tile[TILE_M * IN_CH];          // 8 KB x-tile, shared by 4 waves

  const int m0   = blockIdx.x * TILE_M;
  const int tid  = threadIdx.x;
  const int lane = tid & 31;
  const int n0   = (tid >> 5) * 16;
  const int r    = lane & 15;          // row (A) / col (B/D) within tile
  const int hi   = lane >> 4;          // lane-half selects K pair / M half

  // --- stage x[m0:m0+16, :] into LDS: 512 x float4, 4 async b128 per thread
#if __has_builtin(__builtin_amdgcn_global_load_async_to_lds_b128)
#pragma unroll
  for (int q = 0; q < 4; ++q) {
    const int idx  = tid + q * 128;               // float4 index 0..511
    const int row  = idx >> 5;
    const int colv = idx & 31;
    const float* gp = x + (size_t)(m0 + row) * IN_CH + colv * 4;
    __builtin_amdgcn_global_load_async_to_lds_b128(
        (gbl_v4i_ptr)(uintptr_t)gp,
        (lds_v4i_ptr)(uint32_t)(uintptr_t)(&tile[idx * 4]),
        0, 0);
  }
#if __has_builtin(__builtin_amdgcn_s_wait_asynccnt)
  __builtin_amdgcn_s_wait_asynccnt(0);
#else
  asm volatile("s_wait_asynccnt 0x0" ::: "memory");
#endif
#else
  // fallback: plain vector loads + LDS stores
#pragma unroll
  for (int q = 0; q < 4; ++q) {
    const int idx  = tid + q * 128;
    const int row  = idx >> 5;
    const int colv = idx & 31;
    *(float4*)&tile[idx * 4] =
        *(const float4*)(x + (size_t)(m0 + row) * IN_CH + colv * 4);
  }
#endif
  __syncthreads();

  const float* ar = &tile[(size_t)r * IN_CH];     // A: row m0+r (from LDS)
  const float* wp = W + n0 + r;                   // B: column n0+r (L2-hot)

  v8f acc = {};
#pragma unroll 4
  for (int k = 0; k < IN_CH; k += 4) {
    v2f a, b;
    // A 16x4 f32 layout: v0 = K=0 (lanes0-15) / K=2 (lanes16-31); v1 = K=1 / K=3
    a[0] = ar[k + 2 * hi];
    a[1] = ar[k + 2 * hi + 1];
    // B 4x16 f32 layout mirrors: v0 = K=0/K=2 at N=lane, v1 = K=1/K=3
    b[0] = wp[(size_t)(k + 2 * hi) * OUT_CH];
    b[1] = wp[(size_t)(k + 2 * hi + 1) * OUT_CH];
    acc = __builtin_amdgcn_wmma_f32_16x16x4_f32(false, a, false, b,
                                                (short)0, acc, false, false);
  }
  // D 16x16 f32: VGPR q -> M = q (lanes 0-15) / M = 8+q (lanes 16-31), N = n0+r
  float* hp = h + (size_t)(m0 + 8 * hi) * OUT_CH + n0 + r;
#pragma unroll
  for (int q = 0; q < 8; ++q) hp[(size_t)q * OUT_CH] = acc[q];
}

// ---------------- aggregation ----------------
__global__ void k_agg_init(const float* __restrict__ h, const float* __restrict__ dinv,
                           float* __restrict__ agg, long long total) {
  long long t = (long long)blockIdx.x * blockDim.x + threadIdx.x;
  if (t < total) {
    int i = (int)(t >> 6);
    float d = dinv[i];
    agg[t] = h[t] * d * d;                        // self-loop contribution
  }
}

__global__ void k_agg_edges(const int* __restrict__ ei, const float* __restrict__ h,
                            const float* __restrict__ dinv, float* __restrict__ agg,
                            int E) {
  long long t = (long long)blockIdx.x * blockDim.x + threadIdx.x;
  if (t < (long long)E * OUT_CH) {
    int e = (int)(t >> 6);
    int c = (int)(t & 63);
    int s = ei[e];          // row (source)
    int d = ei[E + e];      // col (target)
    float nrm = dinv[s] * dinv[d];
    atomicAdd(&agg[(size_t)d * OUT_CH + c], h[(size_t)s * OUT_CH + c] * nrm);
  }
}

__global__ void k_relu_bias(float* agg, const float* __restrict__ b, long long total) {
  long long t = (long long)blockIdx.x * blockDim.x + threadIdx.x;
  if (t < total) {
    float v = agg[t] + b[t & 63];
    agg[t] = v > 0.0f ? v : 0.0f;
  }
}

// ---------------- edge classifier: wave per edge ----------------
__global__ __launch_bounds__(256) void k_classify(const int* __restrict__ ei,
                                                  const int* __restrict__ nei,
                                                  const float* __restrict__ z,
                                                  const float* __restrict__ Wc,
                                                  const float* __restrict__ bc,
                                                  float* __restrict__ out, int E) {
  const int wid  = (blockIdx.x * blockDim.x + threadIdx.x) >> 5;
  const int lane = threadIdx.x & 31;
  const int TE = 2 * E;
  if (wid >= TE) return;
  int s, d;
  if (wid < E) { s = ei[wid];       d = ei[E + wid]; }
  else         { s = nei[wid - E];  d = nei[wid]; }   // nei[(wid-E)], nei[E+(wid-E)]

  // lane covers channels 2l,2l+1 of each 64-wide half-feature (float2 = coalesced row)
  const float2 zs = *(const float2*)(z + (size_t)s * OUT_CH + 2 * lane);
  const float2 zd = *(const float2*)(z + (size_t)d * OUT_CH + 2 * lane);
  const float4 ws = *(const float4*)(Wc + 4 * lane);        // rows 2l,2l+1
  const float4 wd = *(const float4*)(Wc + 128 + 4 * lane);  // rows 64+2l,64+2l+1

  float a0 = zs.x * ws.x + zs.y * ws.z + zd.x * wd.x + zd.y * wd.z;
  float a1 = zs.x * ws.y + zs.y * ws.w + zd.x * wd.y + zd.y * wd.w;
#pragma unroll
  for (int m = 16; m >= 1; m >>= 1) {               // wave32 butterfly reduce
    a0 += __shfl_xor(a0, m, 32);
    a1 += __shfl_xor(a1, m, 32);
  }
  if (lane == 0) {
    out[(size_t)wid * 2]     = a0 + bc[0];
    out[(size_t)wid * 2 + 1] = a1 + bc[1];
  }
}

extern "C" void kernel_launch(void* const* d_in, const int* in_sizes, int n_in,
                              void* d_out, int out_size, void* d_ws, size_t ws_size,
                              hipStream_t stream) {
  const float* x   = (const float*)d_in[0];
  const int*   ei  = (const int*)d_in[1];
  const int*   nei = (const int*)d_in[2];
  const float* W   = (const float*)d_in[3];
  const float* b   = (const float*)d_in[4];
  const float* Wc  = (const float*)d_in[5];
  const float* bc  = (const float*)d_in[6];
  float* out = (float*)d_out;

  const int N = in_sizes[0] / IN_CH;
  const int E = in_sizes[1] / 2;

  float* h   = (float*)d_ws;                       // N*64 f32
  float* agg = h + (size_t)N * OUT_CH;             // N*64 f32
  float* deg = agg + (size_t)N * OUT_CH;           // N f32 (becomes dinv in place)

  const long long tn = (long long)N * OUT_CH;
  const long long te = (long long)E * OUT_CH;

  k_init_deg <<<(N + 255) / 256, 256, 0, stream>>>(deg, N);
  k_deg_edges<<<(E + 255) / 256, 256, 0, stream>>>(ei, deg, E);
  k_dinv     <<<(N + 255) / 256, 256, 0, stream>>>(deg, N);

  k_gemm_wmma<<<N / TILE_M, 128, 0, stream>>>(x, W, h);

  k_agg_init <<<(int)((tn + 255) / 256), 256, 0, stream>>>(h, deg, agg, tn);
  k_agg_edges<<<(int)((te + 255) / 256), 256, 0, stream>>>(ei, h, deg, agg, E);
  k_relu_bias<<<(int)((tn + 255) / 256), 256, 0, stream>>>(agg, b, tn);

  const int TE = 2 * E;
  k_classify <<<(int)(((long long)TE * 32 + 255) / 256), 256, 0, stream>>>(
      ei, nei, agg, Wc, bc, out, E);
}